// CPDLoss_14843406975338
// MI455X (gfx1250) — compile-verified
//
#include <hip/hip_runtime.h>
#include <stdint.h>

#define TPB 256
#define MMAX 64

typedef float v2f __attribute__((ext_vector_type(2)));
typedef float v8f __attribute__((ext_vector_type(8)));

// ---- CDNA5 async global->LDS staging -------------------------------------
// Lower 32 bits of a generic LDS pointer are the wave-relative LDS byte
// address (ISA 10.2), which is what the async
// instruction's VDST operand expects.
__device__ __forceinline__ unsigned lds_off(const void* p) {
    return (unsigned)(uintptr_t)p;
}

// Issue per-lane async b32 copies global->LDS, then drain ASYNCcnt.
// Each wave waits for its own asyncs; caller must __syncthreads() after.
__device__ __forceinline__ void async_copy_to_lds(float* ldst, const float* gsrc, int ndw) {
    for (int i = threadIdx.x; i < ndw; i += blockDim.x) {
        unsigned lo = lds_off(ldst + i);
        const float* ga = gsrc + i;
        asm volatile("global_load_async_to_lds_b32 %0, %1, off"
                     :: "v"(lo), "v"(ga) : "memory");
    }
    asm volatile("s_wait_asynccnt 0x0" ::: "memory");
}

// ---- WMMA wave-level reduction (V_WMMA_F32_16X16X4_F32) -------------------
// Selection A-matrix: A[0][k]=[1,0,1,0], A[1][k]=[0,1,0,1], rows>=2 zero.
// A 16x4 layout: lanes 0-15 {VGPR0=K0, VGPR1=K1}, lanes 16-31 {K2, K3}
//   -> a.x = (row==0), a.y = (row==1), identical in both lane halves.
// B 4x16 holds the data: VGPR0 = channel X (K0 lanes 0-15, K2 lanes 16-31),
//                        VGPR1 = channel Y.
// D[0][j] = x_j + x_{j+16}  (D VGPR0, lanes 0-15)
// D[1][j] = y_j + y_{j+16}  (D VGPR1, lanes 0-15)
// Finish with a 4-step width-16 shuffle tree; lane 0 holds the wave sums.
__device__ __forceinline__ void wave_reduce3_wmma(float x, float y, float z,
                                                  float& ox, float& oy, float& oz) {
    const int row = threadIdx.x & 15;
    v2f a; a.x = (row == 0) ? 1.f : 0.f; a.y = (row == 1) ? 1.f : 0.f;
    v2f b1; b1.x = x; b1.y = y;
    v2f b2; b2.x = z; b2.y = 0.f;
    v8f c = {};
    v8f d1 = __builtin_amdgcn_wmma_f32_16x16x4_f32(false, a, false, b1,
                                                   (short)0, c, false, false);
    v8f d2 = __builtin_amdgcn_wmma_f32_16x16x4_f32(false, a, false, b2,
                                                   (short)0, c, false, false);
    float sx = d1[0], sy = d1[1], sz = d2[0];
    #pragma unroll
    for (int off = 8; off > 0; off >>= 1) {
        sx += __shfl_xor(sx, off, 16);
        sy += __shfl_xor(sy, off, 16);
        sz += __shfl_xor(sz, off, 16);
    }
    ox = sx; oy = sy; oz = sz;   // valid on lane 0 of each wave
}

// ---- Kernel 0: zero per-launch accumulators (ws is not re-poisoned) ------
__global__ void init_kernel(unsigned long long* gkeys, int* numPos,
                            float* lossL, float* cePos, float* lossC,
                            int B, int M) {
    int i = blockIdx.x * blockDim.x + threadIdx.x;
    if (i < B * M) gkeys[i] = 0ull;
    if (i < B) { numPos[i] = 0; lossL[i] = 0.f; cePos[i] = 0.f; lossC[i] = 0.f; }
}

// ---- Kernel 1: matching --------------------------------------------------
// Per (b, anchor): max/argmax IoU over M targets (invalid targets -> -1),
// and per-target best-prior argmax over anchors via packed-u64 atomic max.
__global__ void match_kernel(const float* __restrict__ anchors,
                             const float* __restrict__ targets,
                             float* __restrict__ bestOv,
                             int* __restrict__ bestIdx,
                             unsigned long long* __restrict__ gkeys,
                             int A, int M, int bpb) {
    __shared__ float sraw[MMAX * 5];
    __shared__ float sx1[MMAX], sy1[MMAX], sx2[MMAX], sy2[MMAX], sarea[MMAX], slab[MMAX];
    __shared__ unsigned long long skey[MMAX];
    const int tid = threadIdx.x;
    const int b = blockIdx.x / bpb;
    const int a = (blockIdx.x % bpb) * TPB + tid;

    async_copy_to_lds(sraw, targets + (size_t)b * M * 5, M * 5);
    __syncthreads();
    if (tid < M) {
        float x1 = sraw[tid*5+0], y1 = sraw[tid*5+1];
        float x2 = sraw[tid*5+2], y2 = sraw[tid*5+3];
        sx1[tid] = x1; sy1[tid] = y1; sx2[tid] = x2; sy2[tid] = y2;
        sarea[tid] = (x2 - x1) * (y2 - y1);
        slab[tid]  = sraw[tid*5+4];
        skey[tid]  = 0ull;
    }
    __syncthreads();

    const float4 av = ((const float4*)anchors)[a];
    const float ax1 = av.x - av.z * 0.5f, ay1 = av.y - av.w * 0.5f;
    const float ax2 = av.x + av.z * 0.5f, ay2 = av.y + av.w * 0.5f;
    const float areaA = (ax2 - ax1) * (ay2 - ay1);

    float best = -2.0f; int bi = 0;
    for (int m = 0; m < M; ++m) {
        const bool valid = slab[m] > 0.f;     // wave-uniform (from LDS)
        float iou = 0.f, ov;
        if (valid) {
            float lx = fmaxf(sx1[m], ax1), ly = fmaxf(sy1[m], ay1);
            float rx = fminf(sx2[m], ax2), ry = fminf(sy2[m], ay2);
            float w = fmaxf(rx - lx, 0.f), h = fmaxf(ry - ly, 0.f);
            float inter = w * h;
            iou = inter / (sarea[m] + areaA - inter);
            ov = iou;
        } else {
            ov = -1.0f;
        }
        if (ov > best) { best = ov; bi = m; }   // strict > = first-index argmax
        if (valid) {
            // wave32 cross-lane max; leader = lowest lane = smallest anchor idx
            float r = iou;
            #pragma unroll
            for (int off = 16; off > 0; off >>= 1)
                r = fmaxf(r, __shfl_xor(r, off, 32));
            unsigned long long mk = __ballot(iou == r);
            int leader = __ffsll(mk) - 1;
            if ((tid & 31) == leader) {
                unsigned long long key =
                    ((unsigned long long)__float_as_uint(iou) << 32) |
                    (unsigned)(~(unsigned)a);   // bigger key => higher iou, then smaller a
                atomicMax(&skey[m], key);       // ds_max_u64
            }
        }
    }
    const size_t gid = (size_t)b * A + a;
    bestOv[gid]  = best;
    bestIdx[gid] = bi;
    __syncthreads();
    if (tid < M) atomicMax(&gkeys[b * M + tid], skey[tid]);
}

// ---- Kernel 2: forced-match scatter (sequential => last write wins) ------
__global__ void scatter_kernel(const float* __restrict__ targets,
                               const unsigned long long* __restrict__ gkeys,
                               float* __restrict__ bestOv,
                               int* __restrict__ bestIdx, int A, int M) {
    const int b = blockIdx.x;
    if (threadIdx.x != 0) return;
    for (int m = 0; m < M; ++m) {
        float lb = targets[((size_t)b * M + m) * 5 + 4];
        if (lb > 0.f) {
            unsigned a = ~(unsigned)(gkeys[b * M + m] & 0xFFFFFFFFull);
            size_t gid = (size_t)b * A + a;
            bestOv[gid]  = 2.0f;
            bestIdx[gid] = m;
        }
    }
}

// ---- Kernel 3: fused per-anchor loss pass --------------------------------
// Reads bestOv, writes proxy into the SAME buffer (ovProxy).
// Block reduction: WMMA wave-level fold (32->16 lanes, 2 channels per op)
// + width-16 shuffle tree + tiny LDS combine across waves.
__global__ void loss_kernel(const float* __restrict__ anchors,
                            const float* __restrict__ targets,
                            const float* __restrict__ loc_pred,
                            const float* __restrict__ conf_pred,
                            float* __restrict__ ovProxy,
                            const int* __restrict__ bestIdx,
                            float* __restrict__ lossL,
                            float* __restrict__ cePos,
                            int* __restrict__ numPos,
                            int A, int M, int bpb) {
    __shared__ float sraw[MMAX * 5];
    __shared__ float wL[TPB / 32], wC[TPB / 32], wP[TPB / 32];
    const int tid = threadIdx.x;
    const int b = blockIdx.x / bpb;
    const int a = (blockIdx.x % bpb) * TPB + tid;

    async_copy_to_lds(sraw, targets + (size_t)b * M * 5, M * 5);
    __syncthreads();

    const size_t gid = (size_t)b * A + a;
    const float ov = ovProxy[gid];
    const int   mi = bestIdx[gid];
    const float tx1 = sraw[mi*5+0], ty1 = sraw[mi*5+1];
    const float tx2 = sraw[mi*5+2], ty2 = sraw[mi*5+3];
    const float lb  = sraw[mi*5+4];
    const int  conf = (ov < 0.5f) ? 0 : (lb > 0.f ? 1 : 0);
    const bool pos  = conf > 0;

    const float4 av = ((const float4*)anchors)[a];
    const float cx = (tx1 + tx2) * 0.5f, cy = (ty1 + ty2) * 0.5f;
    const float ltx = (cx - av.x) / (0.1f * av.z);
    const float lty = (cy - av.y) / (0.1f * av.w);
    const float2 lp = ((const float2*)loc_pred)[gid];
    const float d0 = fabsf(lp.x - ltx), d1 = fabsf(lp.y - lty);
    const float s0 = d0 < 1.f ? 0.5f * d0 * d0 : d0 - 0.5f;
    const float s1 = d1 < 1.f ? 0.5f * d1 * d1 : d1 - 0.5f;

    const float2 cp = ((const float2*)conf_pred)[gid];
    const float mx  = fmaxf(cp.x, cp.y);
    const float lse = mx + logf(expf(cp.x - mx) + expf(cp.y - mx));
    const float ce  = lse - (conf ? cp.y : cp.x);

    ovProxy[gid] = pos ? 0.0f : ce;   // proxy for hard-negative mining

    // 3-channel block reduction: matrix-pipe wave fold, then LDS combine.
    float sx, sy, sz;
    wave_reduce3_wmma(pos ? (s0 + s1) : 0.f,
                      pos ? ce : 0.f,
                      pos ? 1.f : 0.f, sx, sy, sz);
    const int wid = tid >> 5;
    if ((tid & 31) == 0) { wL[wid] = sx; wC[wid] = sy; wP[wid] = sz; }
    __syncthreads();
    if (tid == 0) {
        float L = 0.f, C = 0.f, P = 0.f;
        #pragma unroll
        for (int w = 0; w < TPB / 32; ++w) { L += wL[w]; C += wC[w]; P += wP[w]; }
        atomicAdd(&lossL[b], L);
        atomicAdd(&cePos[b], C);
        atomicAdd(&numPos[b], (int)P);   // exact: small integer in f32
    }
}

// ---- Kernel 4: exact top-k (hard negative mining) via 3-level radix select
__global__ void select_kernel(const float* __restrict__ proxy,
                              const int* __restrict__ numPos,
                              const float* __restrict__ cePos,
                              float* __restrict__ lossC, int A) {
    __shared__ int hist[2048];
    __shared__ unsigned sbin1, sbin2, sbin3;
    __shared__ int srem1, srem2;
    __shared__ float redf[TPB];
    __shared__ int   redi[TPB];
    const int b = blockIdx.x;
    const int tid = threadIdx.x;
    const unsigned* pbits = (const unsigned*)(proxy + (size_t)b * A);

    const int np = numPos[b];
    int k = 3 * np; { int mx = A - np; if (k > mx) k = mx; }
    if (k <= 0) { if (tid == 0) lossC[b] = cePos[b]; return; }

    // level 1: bits[31:21]
    for (int i = tid; i < 2048; i += TPB) hist[i] = 0;
    __syncthreads();
    for (int i = tid; i < A; i += TPB) atomicAdd(&hist[pbits[i] >> 21], 1);
    __syncthreads();
    if (tid == 0) {
        int rem = k; unsigned bin = 0;
        for (int j = 2047; j >= 0; --j) { if (hist[j] >= rem) { bin = j; break; } rem -= hist[j]; }
        sbin1 = bin; srem1 = rem;
    }
    __syncthreads();
    const unsigned bin1 = sbin1; int kr = srem1;

    // level 2: bits[20:10] within bin1
    for (int i = tid; i < 2048; i += TPB) hist[i] = 0;
    __syncthreads();
    for (int i = tid; i < A; i += TPB) {
        unsigned v = pbits[i];
        if ((v >> 21) == bin1) atomicAdd(&hist[(v >> 10) & 0x7FFu], 1);
    }
    __syncthreads();
    if (tid == 0) {
        int rem = kr; unsigned bin = 0;
        for (int j = 2047; j >= 0; --j) { if (hist[j] >= rem) { bin = j; break; } rem -= hist[j]; }
        sbin2 = bin; srem2 = rem;
    }
    __syncthreads();
    const unsigned bin2 = sbin2; kr = srem2;
    const unsigned pref = (bin1 << 21) | (bin2 << 10);

    // level 3: bits[9:0] within (bin1,bin2)
    for (int i = tid; i < 2048; i += TPB) hist[i] = 0;
    __syncthreads();
    for (int i = tid; i < A; i += TPB) {
        unsigned v = pbits[i];
        if ((v >> 10) == (pref >> 10)) atomicAdd(&hist[v & 0x3FFu], 1);
    }
    __syncthreads();
    if (tid == 0) {
        int rem = kr; unsigned bin = 0;
        for (int j = 1023; j >= 0; --j) { if (hist[j] >= rem) { bin = j; break; } rem -= hist[j]; }
        sbin3 = bin;
    }
    __syncthreads();
    const unsigned tbits = pref | sbin3;     // k-th largest proxy (bit pattern)

    // final: sum of strictly-greater values + tie contribution
    float ssum = 0.f; int cgt = 0;
    for (int i = tid; i < A; i += TPB) {
        unsigned v = pbits[i];
        if (v > tbits) { ssum += __uint_as_float(v); cgt++; }
    }
    redf[tid] = ssum; redi[tid] = cgt;
    __syncthreads();
    for (int s = TPB / 2; s > 0; s >>= 1) {
        if (tid < s) { redf[tid] += redf[tid+s]; redi[tid] += redi[tid+s]; }
        __syncthreads();
    }
    if (tid == 0) {
        float t = __uint_as_float(tbits);
        lossC[b] = cePos[b] + redf[0] + (float)(k - redi[0]) * t;
    }
}

// ---- Kernel 5: deterministic final combine -------------------------------
__global__ void finalize_kernel(const int* numPos, const float* lossL,
                                const float* lossC, float* out, int B) {
    if (blockIdx.x == 0 && threadIdx.x == 0) {
        float L = 0.f, C = 0.f; int P = 0;
        for (int b = 0; b < B; ++b) { L += lossL[b]; C += lossC[b]; P += numPos[b]; }
        float tp = (float)P;
        out[0] = L / tp;
        out[1] = C / tp;
    }
}

// ---- Host launch ---------------------------------------------------------
extern "C" void kernel_launch(void* const* d_in, const int* in_sizes, int n_in,
                              void* d_out, int out_size, void* d_ws, size_t ws_size,
                              hipStream_t stream) {
    const float* loc_pred  = (const float*)d_in[0];
    const float* conf_pred = (const float*)d_in[1];
    const float* anchors   = (const float*)d_in[2];
    const float* targets   = (const float*)d_in[3];

    const int A = in_sizes[2] / 4;                 // anchors: (A,4)
    const int B = in_sizes[0] / (A * 2);           // loc_pred: (B,A,2)
    const int M = in_sizes[3] / (B * 5);           // targets: (B,M,5)
    const int bpb = A / TPB;                       // A divisible by 256

    char* ws = (char*)d_ws;
    auto carve = [&](size_t bytes) { char* p = ws; ws += (bytes + 255) & ~(size_t)255; return p; };
    unsigned long long* gkeys = (unsigned long long*)carve((size_t)B * M * 8);
    float* ovProxy = (float*)carve((size_t)B * A * 4);   // bestOv, then proxy
    int*   bestIdx = (int*)  carve((size_t)B * A * 4);
    int*   numPos  = (int*)  carve((size_t)B * 4);
    float* lossL   = (float*)carve((size_t)B * 4);
    float* cePos   = (float*)carve((size_t)B * 4);
    float* lossC   = (float*)carve((size_t)B * 4);

    init_kernel<<<(B * M + 255) / 256, 256, 0, stream>>>(gkeys, numPos, lossL, cePos, lossC, B, M);
    match_kernel<<<B * bpb, TPB, 0, stream>>>(anchors, targets, ovProxy, bestIdx, gkeys, A, M, bpb);
    scatter_kernel<<<B, 32, 0, stream>>>(targets, gkeys, ovProxy, bestIdx, A, M);
    loss_kernel<<<B * bpb, TPB, 0, stream>>>(anchors, targets, loc_pred, conf_pred,
                                             ovProxy, bestIdx, lossL, cePos, numPos, A, M, bpb);
    select_kernel<<<B, TPB, 0, stream>>>(ovProxy, numPos, cePos, lossC, A);
    finalize_kernel<<<1, 32, 0, stream>>>(numPos, lossL, lossC, (float*)d_out, B);
}